// QRMSNorm_40243843564244
// MI455X (gfx1250) — compile-verified
//
#include <hip/hip_runtime.h>
#include <stdint.h>

// Fused residual-add (fp32) + RMSNorm + dynamic per-token int8 quant.
// B*S = 16384 rows, H = 4096. Pure streaming / bandwidth-bound kernel:
//   reads : x(fp16) + residual(fp16) + weight(fp16, L2-resident)
//   writes: y_q(as float) + residual_out(fp32) + per_token_scale(fp32)
// CDNA5 path: GLOBAL_LOAD_ASYNC_TO_LDS_B128 stages both fp16 tiles into LDS
// (ASYNCcnt-tracked), s_wait_asynccnt 0, then ds_load_b128 reads. Each thread
// async-loads exactly the bytes it later consumes, so no barrier is needed
// for the tiles. Single fused pass computes both sum(xr^2) and max|xr*w|
// (valid because rstd is a positive per-token scalar), so only ONE block
// reduction is required; t = xr*w is held in VGPRs for the quantize pass.

#define HDIM      4096
#define NTHREADS  256
#define EPT       (HDIM / NTHREADS)   // 16 elements per thread

typedef _Float16 h8 __attribute__((ext_vector_type(8)));   // 16 bytes
typedef float    f4 __attribute__((ext_vector_type(4)));   // 16 bytes

__launch_bounds__(NTHREADS)
__global__ void qrmsnorm_fused_kernel(const _Float16* __restrict__ x,
                                      const _Float16* __restrict__ r,
                                      const _Float16* __restrict__ w,
                                      float* __restrict__ out_q,
                                      float* __restrict__ out_res,
                                      float* __restrict__ out_scale)
{
    __shared__ __align__(16) _Float16 lx[HDIM];           // 8 KiB x tile
    __shared__ __align__(16) _Float16 lr[HDIM];           // 8 KiB residual tile
    __shared__ float red_s[NTHREADS / 32];                // per-wave sumsq
    __shared__ float red_m[NTHREADS / 32];                // per-wave amax

    const int    tid     = threadIdx.x;
    const int    row     = blockIdx.x;
    const size_t ebase   = (size_t)row * HDIM + (size_t)tid * EPT;

    // ---- async global -> LDS staging (gfx1250 ASYNCcnt path) ----
    // Each thread moves its own 32 bytes of x and 32 bytes of residual.
    // IOFFSET is added to BOTH the global and LDS address, so one base pair
    // covers both 16-byte chunks.
    const uint8_t* gx = (const uint8_t*)(x + ebase);
    const uint8_t* gr = (const uint8_t*)(r + ebase);
    // LDS pointers: low 32 bits of the flat address ARE the LDS byte offset
    // (shared aperture lives entirely in addr[63:32] on CDNA5).
    uint32_t lox = (uint32_t)(uintptr_t)(lx + tid * EPT);
    uint32_t lorr = (uint32_t)(uintptr_t)(lr + tid * EPT);

    asm volatile(
        "global_load_async_to_lds_b128 %0, %2, off\n\t"
        "global_load_async_to_lds_b128 %0, %2, off offset:16\n\t"
        "global_load_async_to_lds_b128 %1, %3, off\n\t"
        "global_load_async_to_lds_b128 %1, %3, off offset:16"
        :
        : "v"(lox), "v"(lorr), "v"(gx), "v"(gr)
        : "memory");
    asm volatile("s_wait_asynccnt 0x0" ::: "memory");

    // ---- pass 1: xr = x + r (fp32); store residual_out; accumulate
    //      sumsq = sum(xr^2) and amax = max|xr * w| in the SAME pass ----
    const h8* lxp = (const h8*)(lx + tid * EPT);   // ds_load_b128
    const h8* lrp = (const h8*)(lr + tid * EPT);
    const h8* wp  = (const h8*)(w + (size_t)tid * EPT);  // global_load_b128, L2-hot

    float  t[EPT];
    float  sumsq = 0.0f;
    float  amax  = 0.0f;
    float* resout = out_res + ebase;

    #pragma unroll
    for (int half = 0; half < 2; ++half) {
        h8 hx = lxp[half];
        h8 hr = lrp[half];
        h8 hw = wp[half];
        f4 r0, r1;
        #pragma unroll
        for (int i = 0; i < 8; ++i) {
            float xr = (float)hx[i] + (float)hr[i];
            sumsq = fmaf(xr, xr, sumsq);
            float ti = xr * (float)hw[i];
            t[half * 8 + i] = ti;
            amax = fmaxf(amax, fabsf(ti));
            if (i < 4) r0[i] = xr; else r1[i - 4] = xr;
        }
        *(f4*)(resout + half * 8)     = r0;   // global_store_b128
        *(f4*)(resout + half * 8 + 4) = r1;
    }

    // ---- block reduction: wave32 shfl-xor tree, then 8-wave LDS combine ----
    #pragma unroll
    for (int m = 16; m >= 1; m >>= 1) {
        sumsq += __shfl_xor(sumsq, m, 32);
        amax   = fmaxf(amax, __shfl_xor(amax, m, 32));
    }
    const int wave = tid >> 5;
    if ((tid & 31) == 0) { red_s[wave] = sumsq; red_m[wave] = amax; }
    __syncthreads();

    float ssum = 0.0f, smax = 0.0f;
    #pragma unroll
    for (int i = 0; i < NTHREADS / 32; ++i) {   // all threads redundantly:
        ssum += red_s[i];                       // deterministic, no 2nd barrier
        smax  = fmaxf(smax, red_m[i]);
    }

    const float rstd = rsqrtf(ssum * (1.0f / (float)HDIM) + 1e-5f);
    const float pts  = smax * rstd * (1.0f / 127.0f);   // returned (unclamped)
    const float safe = fmaxf(pts, 1e-8f);
    const float qmul = rstd / safe;                      // y/safe = t * qmul

    if (tid == 0) out_scale[row] = pts;

    // ---- pass 2: quantize from registers (RNE round == jnp.round) ----
    float* qout = out_q + ebase;
    #pragma unroll
    for (int v = 0; v < 4; ++v) {
        f4 q;
        #pragma unroll
        for (int i = 0; i < 4; ++i) {
            float qq = rintf(t[v * 4 + i] * qmul);
            q[i] = fminf(fmaxf(qq, -128.0f), 127.0f);
        }
        *(f4*)(qout + v * 4) = q;   // global_store_b128
    }
}

extern "C" void kernel_launch(void* const* d_in, const int* in_sizes, int n_in,
                              void* d_out, int out_size, void* d_ws, size_t ws_size,
                              hipStream_t stream) {
    const _Float16* x = (const _Float16*)d_in[0];   // [B,S,H] fp16
    const _Float16* r = (const _Float16*)d_in[1];   // [B,S,H] fp16
    const _Float16* w = (const _Float16*)d_in[2];   // [H]     fp16

    const size_t n    = (size_t)in_sizes[0];        // B*S*H
    const int    rows = (int)(n / HDIM);            // B*S tokens

    // Tuple outputs concatenated flat in return order, non-bf16 -> float:
    float* out_q     = (float*)d_out;               // y_q values (as float)
    float* out_res   = out_q + n;                   // residual_out fp32
    float* out_scale = out_res + n;                 // per-token scale fp32

    qrmsnorm_fused_kernel<<<rows, NTHREADS, 0, stream>>>(x, r, w,
                                                         out_q, out_res, out_scale);
}